// MemN2NDialog_28544352649953
// MI455X (gfx1250) — compile-verified
//
#include <hip/hip_runtime.h>

typedef __attribute__((ext_vector_type(2))) float v2f;
typedef __attribute__((ext_vector_type(4))) float v4f;
typedef __attribute__((ext_vector_type(8))) float v8f;

#define B_ 64
#define M_ 512
#define S_ 1024
#define E_ 128
#define C_ 10000

// ---------------- Kernel 1a: partial reduce of stories over m ----------------
// grid (B, 8 chunks), block 256. Each lane owns one float4 column (512B/wave
// contiguous per m-step), non-temporal b128 loads: stories is read-once.
__global__ void k_stories_partial(const float* __restrict__ stories,
                                  float* __restrict__ partial) {
  int s4 = threadIdx.x;                      // 0..255 -> floats 4*s4..4*s4+3
  int b  = blockIdx.x;
  int mc = blockIdx.y;                       // 0..7, 64 m's each
  const float* p = stories + ((size_t)b * M_ + (size_t)mc * 64) * S_ + 4 * s4;
  v4f acc = {0.f, 0.f, 0.f, 0.f};
#pragma unroll 8
  for (int m = 0; m < 64; ++m) {
    v4f v = __builtin_nontemporal_load((const v4f*)(p + (size_t)m * S_));
    acc += v;
  }
  *(v4f*)(partial + ((size_t)mc * B_ + b) * S_ + 4 * s4) = acc;
}

// ---------------- Kernel 1b: combine 8 partials -> ssum[B,S] ----------------
// grid B, block 256, float4 per lane.
__global__ void k_stories_final(const float* __restrict__ partial,
                                float* __restrict__ ssum) {
  int s4 = threadIdx.x;
  int b  = blockIdx.x;
  v4f acc = {0.f, 0.f, 0.f, 0.f};
#pragma unroll
  for (int mc = 0; mc < 8; ++mc)
    acc += *(const v4f*)(partial + ((size_t)mc * B_ + b) * S_ + 4 * s4);
  *(v4f*)(ssum + (size_t)b * S_ + 4 * s4) = acc;
}

// ---------------- f32 WMMA 16x16 tile helpers ----------------
// out[i][j] = sum_k A[i*lda + k] * Bt[j*ldb + k]   (i.e. A * Bt^T, both row-major)
// A 16x4 layout: lane l holds row (l&15), K = (l>>4)*2 .. +1  -> one float2.
// B  4x16 layout: lane n holds col (n&15), same K split        -> one float2.
__device__ __forceinline__ v8f wmma_tile_f32(const float* __restrict__ A, int lda,
                                             const float* __restrict__ Bt, int ldb,
                                             int K) {
  int lane = threadIdx.x & 31;
  int r  = lane & 15;
  int kh = (lane >> 4) << 1;   // 0 or 2
  v8f acc = {0.f, 0.f, 0.f, 0.f, 0.f, 0.f, 0.f, 0.f};
#pragma unroll 4
  for (int k = 0; k < K; k += 4) {
    v2f a = *(const v2f*)(A  + (size_t)r * lda + k + kh);
    v2f b = *(const v2f*)(Bt + (size_t)r * ldb + k + kh);
    acc = __builtin_amdgcn_wmma_f32_16x16x4_f32(false, a, false, b, (short)0, acc,
                                                false, false);
  }
  return acc;
}

// C/D layout: VGPR r -> M = r (lanes 0-15) or M = 8+r (lanes 16-31); N = lane&15.
__device__ __forceinline__ void store_tile_f32(v8f acc, float* __restrict__ out,
                                               int ldo, int m0, int n0) {
  int lane = threadIdx.x & 31;
  int n  = lane & 15;
  int mo = (lane >> 4) << 3;   // 0 or 8
#pragma unroll
  for (int r = 0; r < 8; ++r)
    out[(size_t)(m0 + mo + r) * ldo + (n0 + n)] = acc[r];
}

// ---------------- Kernel 2: u0 = queries*W1^T, o = ssum*W2^T --------------
// grid 16 blocks x 128 threads = 64 waves; 32 tiles per GEMM (4 row x 8 col).
__global__ void k_proj(const float* __restrict__ queries,
                       const float* __restrict__ W1,
                       const float* __restrict__ ssum,
                       const float* __restrict__ W2,
                       float* __restrict__ u, float* __restrict__ o) {
  int gw    = blockIdx.x * 4 + (threadIdx.x >> 5);  // 0..63
  int which = gw >> 5;                              // 0: u, 1: o
  int t  = gw & 31;
  int bt = t >> 3;                                  // 0..3
  int et = t & 7;                                   // 0..7
  const float* A  = (which ? ssum : queries) + (size_t)bt * 16 * S_;
  const float* Bt = (which ? W2   : W1)      + (size_t)et * 16 * S_;
  float* O = which ? o : u;
  v8f acc = wmma_tile_f32(A, S_, Bt, S_, S_);
  store_tile_f32(acc, O, E_, bt * 16, et * 16);
}

// ---------------- Kernel 3: 3 hops, row-independent --------------------------
// u <- (o + u*W3^T) / max(||.||2, eps), repeated 3x.  One block per row b.
__global__ void k_hops(const float* __restrict__ W3,
                       const float* __restrict__ o,
                       float* __restrict__ u) {
  __shared__ float su[E_];
  __shared__ float red[E_];
  int b = blockIdx.x;
  int e = threadIdx.x;                // 0..127
  su[e] = u[(size_t)b * E_ + e];
  float oe = o[(size_t)b * E_ + e];
  const float* w = W3 + (size_t)e * E_;
  for (int hop = 0; hop < 3; ++hop) {
    __syncthreads();
    float acc = oe;
#pragma unroll 4
    for (int k = 0; k < E_; ++k) acc = fmaf(su[k], w[k], acc);
    red[e] = acc * acc;
    __syncthreads();
    for (int sft = E_ >> 1; sft > 0; sft >>= 1) {
      if (e < sft) red[e] += red[e + sft];
      __syncthreads();
    }
    float nrm = fmaxf(sqrtf(red[0]), 1e-12f);
    __syncthreads();
    su[e] = acc / nrm;
  }
  __syncthreads();
  u[(size_t)b * E_ + e] = su[e];
}

// ---------------- Kernel 4: logits = u * W4^T [64,10000] --------------------
// grid 625 blocks x 128 threads; wave w handles row-tile w, col-tile blockIdx.x.
__global__ void k_logits(const float* __restrict__ u,
                         const float* __restrict__ W4,
                         float* __restrict__ logits) {
  int bt = threadIdx.x >> 5;                        // 0..3
  int ct = blockIdx.x;                              // 0..624
  const float* A  = u  + (size_t)bt * 16 * E_;
  const float* Bt = W4 + (size_t)ct * 16 * E_;
  v8f acc = wmma_tile_f32(A, E_, Bt, E_, E_);
  store_tile_f32(acc, logits, C_, bt * 16, ct * 16);
}

// ---------------- Kernel 5: row softmax over C=10000 ------------------------
__global__ void k_softmax(const float* __restrict__ logits,
                          float* __restrict__ out) {
  __shared__ float red[256];
  int b = blockIdx.x, t = threadIdx.x;
  const float* x = logits + (size_t)b * C_;
  float m = -3.402823466e38f;
  for (int i = t; i < C_; i += 256) m = fmaxf(m, x[i]);
  red[t] = m; __syncthreads();
  for (int s = 128; s > 0; s >>= 1) {
    if (t < s) red[t] = fmaxf(red[t], red[t + s]);
    __syncthreads();
  }
  m = red[0]; __syncthreads();
  float sum = 0.f;
  for (int i = t; i < C_; i += 256) sum += expf(x[i] - m);
  red[t] = sum; __syncthreads();
  for (int s = 128; s > 0; s >>= 1) {
    if (t < s) red[t] += red[t + s];
    __syncthreads();
  }
  float inv = 1.f / red[0];
  for (int i = t; i < C_; i += 256) out[(size_t)b * C_ + i] = expf(x[i] - m) * inv;
}

extern "C" void kernel_launch(void* const* d_in, const int* in_sizes, int n_in,
                              void* d_out, int out_size, void* d_ws, size_t ws_size,
                              hipStream_t stream) {
  const float* stories = (const float*)d_in[0];
  const float* queries = (const float*)d_in[1];
  const float* W1 = (const float*)d_in[2];
  const float* W2 = (const float*)d_in[3];
  const float* W3 = (const float*)d_in[4];
  const float* W4 = (const float*)d_in[5];

  float* ws = (float*)d_ws;
  // Region 0: partials (8*B*S = 524288 f) live only in k1a/k1b, then reused
  //           for logits (B*C = 640000 f) -> size 640000 f.
  float* partial = ws;
  float* logits  = ws;
  float* ssum = ws + 640000;            // B*S  = 65536 f
  float* u    = ssum + (size_t)B_ * S_; // B*E  = 8192 f
  float* o    = u + (size_t)B_ * E_;    // B*E  = 8192 f
  float* outp = (float*)d_out;

  k_stories_partial<<<dim3(B_, 8), 256, 0, stream>>>(stories, partial);
  k_stories_final <<<B_,           256, 0, stream>>>(partial, ssum);
  k_proj   <<<16,       128, 0, stream>>>(queries, W1, ssum, W2, u, o);
  k_hops   <<<B_,       E_,  0, stream>>>(W3, o, u);
  k_logits <<<C_ / 16,  128, 0, stream>>>(u, W4, logits);
  k_softmax<<<B_,       256, 0, stream>>>(logits, outp);
}